// MyModel_61933428409195
// MI455X (gfx1250) — compile-verified
//
#include <hip/hip_runtime.h>
#include <stdint.h>

// ---------------------------------------------------------------------------
// Problem constants (reference: prediction [B, 25200, 85], conf > 0.25,
// first 300 passing rows per image in original order, rest zeroed).
// ---------------------------------------------------------------------------
#define N_ANCH 25200
#define NCH    85
#define MAXD   300
#define CONF_T 0.25f
#define BLK    1024          // 32 wave32 waves per block
#define STAGE  75            // output slots handled per gather block (4 stages)
#define ROWB   (NCH * 4)     // 340 bytes per row

typedef unsigned int v4u __attribute__((ext_vector_type(4)));
typedef int          v8i __attribute__((ext_vector_type(8)));
typedef int          v4i __attribute__((ext_vector_type(4)));

#if defined(__has_builtin)
#if __has_builtin(__builtin_amdgcn_tensor_load_to_lds) && \
    __has_builtin(__builtin_amdgcn_tensor_store_from_lds)
#define HAVE_TDM 1
#endif
#endif

// ---------------------------------------------------------------------------
// Kernel 1: ordered stream compaction of passing row indices.
// One block per image. Ballot-based block scan over 1024-row chunks with a
// uniform early exit (expected after the FIRST chunk: ~75% pass rate).
// Produces idx[b][0..299] (strictly increasing) and cnt[b] = min(#pass, 300).
// ---------------------------------------------------------------------------
__global__ __launch_bounds__(BLK) void nms_compact_kernel(
    const float* __restrict__ pred, int* __restrict__ out_idx,
    int* __restrict__ out_cnt) {
  const int b    = blockIdx.x;
  const int tid  = threadIdx.x;
  const int lane = tid & 31;
  const int wv   = tid >> 5;

  __shared__ int s_idx[MAXD];
  __shared__ int s_wcnt[BLK / 32];
  __shared__ int s_base;

  if (tid == 0) s_base = 0;
  __syncthreads();

  const float* base = pred + (size_t)b * N_ANCH * NCH;

  for (int chunk = 0; chunk < N_ANCH; chunk += BLK) {
    const int i = chunk + tid;
    bool pass = false;
    if (i < N_ANCH) {
      pass = base[(size_t)i * NCH + 4] > CONF_T;
      // Hide the 340B-strided conf-column latency for the (rare) next chunk.
      if (i + BLK < N_ANCH)
        __builtin_prefetch(&base[(size_t)(i + BLK) * NCH + 4], 0, 0);
    }
    const unsigned m = __builtin_amdgcn_ballot_w32(pass);   // wave32 ballot
    if (lane == 0) s_wcnt[wv] = __popc(m);
    __syncthreads();

    int woff = 0;
    for (int w = 0; w < wv; ++w) woff += s_wcnt[w];
    const int rank = s_base + woff + __popc(m & ((1u << lane) - 1u));
    if (pass && rank < MAXD) s_idx[rank] = i;
    __syncthreads();

    if (tid == 0) {
      int tot = 0;
      for (int w = 0; w < BLK / 32; ++w) tot += s_wcnt[w];
      s_base += tot;
    }
    __syncthreads();
    if (s_base >= MAXD) break;  // uniform early exit
  }

  int K = s_base;
  if (K > MAXD) K = MAXD;
  if (tid == 0) out_cnt[b] = K;
  for (int j = tid; j < MAXD; j += BLK)
    out_idx[b * MAXD + j] = (j < K) ? s_idx[j] : 0;
}

// ---------------------------------------------------------------------------
// Kernel 2: pure-TDM data path.
//   gather-mode tensor_load_to_lds (16 rows/op, 16-bit indices) -> LDS,
//   zero-fill LDS tail for invalid slots (disjoint region, overlaps the DMA),
//   s_wait_tensorcnt 0, then one tensor_store_from_lds of the whole stage.
// Block (s, b) handles 75 output slots. Indices are a strictly increasing
// prefix, matching TDM gather-mode OOB rules; 25200 < 65536 so 16-bit fits.
// ---------------------------------------------------------------------------
__global__ __launch_bounds__(BLK) void nms_gather_kernel(
    const float* __restrict__ pred, const int* __restrict__ idx,
    const int* __restrict__ cnt, float* __restrict__ out) {
  const int s   = blockIdx.x;   // stage 0..3
  const int b   = blockIdx.y;   // image
  const int tid = threadIdx.x;

  __shared__ float s_rows[STAGE * NCH];   // 75 rows * 340B = 25.5 KB LDS

  const int K     = cnt[b];
  const int slot0 = s * STAGE;
  const float* base = pred + (size_t)b * N_ANCH * NCH;

  int Kstage = K - slot0;               // valid slots in this stage
  if (Kstage < 0) Kstage = 0;
  if (Kstage > STAGE) Kstage = STAGE;

#if defined(HAVE_TDM)
  const uint32_t lds0 = (uint32_t)(uintptr_t)&s_rows[0];  // LDS byte offset

  if (tid < 32) {  // wave 0 issues the tensor DMA ops (EXEC-independent)
    const uint64_t gaddr = (uint64_t)(uintptr_t)base;

    for (int g = 0; g < STAGE; g += 16) {           // up to 16 rows per TDM op
      int n = Kstage - g;
      if (n <= 0) break;
      if (n > 16) n = 16;

      // Pack up to 16 strictly-increasing 16-bit row indices (D# groups 2/3).
      uint32_t p[8];
      for (int d = 0; d < 8; ++d) {
        const int i0 = slot0 + g + 2 * d;
        uint32_t lo = (2 * d     < n) ? ((uint32_t)idx[b * MAXD + i0]     & 0xFFFFu) : 0u;
        uint32_t hi = (2 * d + 1 < n) ? ((uint32_t)idx[b * MAXD + i0 + 1] & 0xFFFFu) : 0u;
        p[d] = lo | (hi << 16);
      }

      // ---- D# group 0 (ISA §8.3): count=1, gather_mode=1, 16-bit idx ----
      v4u g0;
      g0[0] = 0x80000001u;                                  // count=1 | gather_mode
      g0[1] = lds0 + (uint32_t)g * ROWB;                    // lds_addr
      g0[2] = (uint32_t)(gaddr & 0xFFFFFFFFu);              // global_addr[31:0]
      g0[3] = (uint32_t)((gaddr >> 32) & 0x01FFFFFFu)       // global_addr[56:32]
              | (2u << 30);                                 // type=2 ("image")

      // ---- D# group 1 (ISA §8.4) ----
      v8i g1;
      g1[0] = (int)(2u << 16);                  // workgroup_mask=0, data_size=4B
      g1[1] = (int)(85u << 16);                 // tensor_dim0[15:0]=85
      g1[2] = (int)((uint32_t)N_ANCH << 16);    // tensor_dim1[15:0]=25200
      g1[3] = (int)(85u << 16);                 // tile_dim0=85
      g1[4] = (int)(uint32_t)n;                 // tile_dim1 = #valid indices
      g1[5] = 85;                               // tensor_dim0_stride = 85 elems
      g1[6] = 0;
      g1[7] = 0;

      // ---- D# groups 2/3: gather row indices ----
      v4i g2, g3;
      g2[0] = (int)p[0]; g2[1] = (int)p[1]; g2[2] = (int)p[2]; g2[3] = (int)p[3];
      g3[0] = (int)p[4]; g3[1] = (int)p[5]; g3[2] = (int)p[6]; g3[3] = (int)p[7];

#if defined(__clang_major__) && (__clang_major__ >= 23)
      v8i gpad = (v8i)(0);
      __builtin_amdgcn_tensor_load_to_lds(g0, g1, g2, g3, gpad, 0);
#else
      __builtin_amdgcn_tensor_load_to_lds(g0, g1, g2, g3, 0);
#endif
    }
  }

  // Zero-fill the invalid tail [Kstage*85, STAGE*85): byte-disjoint from the
  // region the TDM gathers are writing, so this overlaps with the DMA.
  for (int e = Kstage * NCH + tid; e < STAGE * NCH; e += BLK) s_rows[e] = 0.0f;
  __syncthreads();   // zero-fill visible; wave 0 still owns the tensor ops

  if (tid < 32) {
    __builtin_amdgcn_s_wait_tensorcnt(0);   // gathers done before store reads

    // One contiguous 1-D tile store: stage (6375 floats) -> output slab.
    const uint64_t oaddr =
        (uint64_t)(uintptr_t)(out + (size_t)b * MAXD * NCH +
                              (size_t)slot0 * NCH);
    v4u g0;
    g0[0] = 0x00000001u;                                  // count=1, no gather
    g0[1] = lds0;                                         // lds_addr
    g0[2] = (uint32_t)(oaddr & 0xFFFFFFFFu);              // global_addr[31:0]
    g0[3] = (uint32_t)((oaddr >> 32) & 0x01FFFFFFu) | (2u << 30);  // type=2

    v8i g1;
    g1[0] = (int)(2u << 16);                    // data_size = 4B
    g1[1] = (int)((STAGE * NCH & 0xFFFFu) << 16);  // tensor_dim0[15:0]=6375
    g1[2] = (int)(1u << 16);                    // tensor_dim0 hi=0; tensor_dim1=1
    g1[3] = (int)((uint32_t)(STAGE * NCH) << 16);  // tile_dim0 = 6375
    g1[4] = 0;                                  // tile_dim1/2 unused (1-D tile)
    g1[5] = STAGE * NCH;                        // tensor_dim0_stride
    g1[6] = 0;
    g1[7] = 0;

    v4i g2 = (v4i)(0), g3 = (v4i)(0);

#if defined(__clang_major__) && (__clang_major__ >= 23)
    v8i gpad = (v8i)(0);
    __builtin_amdgcn_tensor_store_from_lds(g0, g1, g2, g3, gpad, 0);
#else
    __builtin_amdgcn_tensor_store_from_lds(g0, g1, g2, g3, 0);
#endif
    __builtin_amdgcn_s_wait_tensorcnt(0);   // store issued & tracked to zero
  }
#else
  // Fallback: cooperative per-thread gather into LDS, then plain write-out.
  for (int e = tid; e < STAGE * NCH; e += BLK) {
    const int slot = slot0 + e / NCH;
    if (slot < K) {
      const int r = idx[b * MAXD + slot];
      s_rows[e] = base[(size_t)r * NCH + (e % NCH)];
    }
  }
  __syncthreads();
  float* outp = out + (size_t)b * MAXD * NCH + (size_t)slot0 * NCH;
  for (int e = tid; e < STAGE * NCH; e += BLK) {
    const int slot = slot0 + e / NCH;
    outp[e] = (slot < K) ? s_rows[e] : 0.0f;
  }
#endif
}

// ---------------------------------------------------------------------------
// Host launch.
// ---------------------------------------------------------------------------
extern "C" void kernel_launch(void* const* d_in, const int* in_sizes, int n_in,
                              void* d_out, int out_size, void* d_ws,
                              size_t ws_size, hipStream_t stream) {
  (void)n_in; (void)out_size; (void)ws_size;
  const float* pred = (const float*)d_in[0];
  float* out = (float*)d_out;

  const int B = in_sizes[0] / (N_ANCH * NCH);

  int* ws_idx = (int*)d_ws;                 // B * 300 ints
  int* ws_cnt = ws_idx + (size_t)B * MAXD;  // B ints

  nms_compact_kernel<<<dim3(B), dim3(BLK), 0, stream>>>(pred, ws_idx, ws_cnt);
  nms_gather_kernel<<<dim3(4, B), dim3(BLK), 0, stream>>>(pred, ws_idx, ws_cnt,
                                                          out);
}